// DPLayer_37048387896036
// MI455X (gfx1250) — compile-verified
//
#include <hip/hip_runtime.h>
#include <stdint.h>

// ---------------------------------------------------------------------------
// Grid longest-path DP:  v[i,j] = max(v[i-1,j], v[i,j-1]) + |x[i,j]|, v[0,0]=0
// B=4096 images of 64x64 f32; out[b] = v[63,63].
//
// Memory-bound (64 MiB @ 23.3 TB/s ~= 2.9 us floor; ~1.3 flop/byte). The
// recurrence lives in the (max,+) semiring -> not expressible on the WMMA
// XDL, so the gfx1250 features that pay are:
//   * global_load_async_to_lds_b128 (ASYNCcnt) quad-buffered pair staging
//   * DPP row_shr Kogge-Stone scans (VALU-only; deps via s_delay_alu,
//     no ds_bpermute / s_wait_dscnt round trips)
// One wave32 per image (4096 waves -> full occupancy). Lane l owns columns
// (2l, 2l+1); each row solved in closed form v = c + cummax(a - c).
// Steady-state loop is branch-free; DMA drain is peeled into an epilogue.
// ---------------------------------------------------------------------------

#define DP_J 64

// ---- DPP helpers: shift-up by D within 16-lane DPP rows --------------------
// bound_ctrl:1 -> invalid source lanes read 0 (identity for +)
template <int D>
__device__ __forceinline__ float dpp_shr_or_zero(float x) {
    return __int_as_float(__builtin_amdgcn_update_dpp(
        0, __float_as_int(x), 0x110 | D, 0xf, 0xf, true));
}
// old = src -> invalid source lanes keep self (no-op for max(t,t))
template <int D>
__device__ __forceinline__ float dpp_shr_or_self(float x) {
    int xi = __float_as_int(x);
    return __int_as_float(__builtin_amdgcn_update_dpp(
        xi, xi, 0x110 | D, 0xf, 0xf, false));
}

__device__ __forceinline__ float bcast_lane15(float x) {
    return __int_as_float(__builtin_amdgcn_readlane(__float_as_int(x), 15));
}

// inclusive +-scan of one value per lane across the wave32
__device__ __forceinline__ float scan_sum32(float t, bool hi) {
    t += dpp_shr_or_zero<1>(t);
    t += dpp_shr_or_zero<2>(t);
    t += dpp_shr_or_zero<4>(t);
    t += dpp_shr_or_zero<8>(t);
    float u = bcast_lane15(t);          // inclusive sum of lanes 0..15
    if (hi) t += u;
    return t;
}
// inclusive max-scan; also returns lane-15 inclusive max (for lane-16 patch)
__device__ __forceinline__ float scan_max32(float t, bool hi, float& lo15) {
    t = fmaxf(t, dpp_shr_or_self<1>(t));
    t = fmaxf(t, dpp_shr_or_self<2>(t));
    t = fmaxf(t, dpp_shr_or_self<4>(t));
    t = fmaxf(t, dpp_shr_or_self<8>(t));
    float u = bcast_lane15(t);
    lo15 = u;
    if (hi) t = fmaxf(t, u);
    return t;
}

// ---- async DMA helpers -----------------------------------------------------
__device__ __forceinline__ uint32_t lds_byte_addr(const void* p) {
    return (uint32_t)(uintptr_t)p;      // low 32 bits of generic LDS ptr
}

// stage one row PAIR (512B): 32 lanes x b128, tracked by ASYNCcnt
__device__ __forceinline__ void stage_pair(const float* gsrc, uint32_t ldsdst) {
    uint64_t ga = (uint64_t)(uintptr_t)gsrc;
    asm volatile("global_load_async_to_lds_b128 %0, %1, off"
                 :: "v"(ldsdst), "v"(ga)
                 : "memory");
}

template <int N>
__device__ __forceinline__ void wait_async() {
    asm volatile("s_wait_asynccnt %0" :: "n"(N) : "memory");
}

// ---- per-row pieces --------------------------------------------------------
// Part 1 (independent of v): weights and prefix sums c[j] = sum_{m=1..j} w[m]
struct RowC { float w0, w1, c0, c1; };

__device__ __forceinline__ RowC row_prefix(float2 w, int lane, bool hi) {
    RowC r;
    r.w0 = fabsf(w.x);
    r.w1 = fabsf(w.y);
    float z0 = (lane == 0) ? 0.f : r.w0;    // column-0 weight excluded from c
    float ps = z0 + w.y * 0.f + (z0 - z0) + fabsf(w.y); // (kept simple below)
    ps = z0 + r.w1;                         // pair total
    float t  = scan_sum32(ps, hi);
    float e  = t - ps;                      // exclusive lane offset
    r.c0 = e + z0;
    r.c1 = e + ps;
    return r;
}

// Part 2 (depends on v): v = c + cummax((v_prev + w) - c)
__device__ __forceinline__ void row_combine(const RowC& r, int lane, bool hi,
                                            float& v0, float& v1) {
    float a0 = v0 + r.w0, a1 = v1 + r.w1;
    float d0 = a0 - r.c0, d1 = a1 - r.c1;
    float pm = fmaxf(d0, d1);               // pair inclusive max
    float u;
    float tm = scan_max32(pm, hi, u);       // inclusive cummax per lane
    float ex = dpp_shr_or_self<1>(tm);      // exclusive; wrong at lanes 0, 16
    ex = (lane == 16) ? u : ex;             // lane 16: inclusive max lanes 0..15
    float m0 = (lane == 0) ? d0 : fmaxf(ex, d0);
    float m1 = (lane == 0) ? pm : fmaxf(ex, pm);
    v0 = r.c0 + m0;
    v1 = r.c1 + m1;
}

// Process one staged pair of rows; prefixes computed up front so both
// independent sum-scan DPP chains can interleave before the serial max-scans.
__device__ __forceinline__ void dp_pair(const float* base, int lane, bool hi,
                                        float& v0, float& v1) {
    float2 wa = *(const float2*)(base + 2 * lane);          // row 2p
    float2 wb = *(const float2*)(base + DP_J + 2 * lane);   // row 2p+1
    RowC ra = row_prefix(wa, lane, hi);
    RowC rb = row_prefix(wb, lane, hi);
    row_combine(ra, lane, hi, v0, v1);
    row_combine(rb, lane, hi, v0, v1);
}

// ---------------------------------------------------------------------------
__global__ __launch_bounds__(32)
void grid_dp_kernel(const float* __restrict__ img, float* __restrict__ out) {
    const int  lane = (int)threadIdx.x;     // 0..31 (wave32)
    const bool hi   = lane >= 16;
    const int  b    = (int)blockIdx.x;
    const float* gbase = img + (size_t)b * (64 * DP_J);

    // quad-buffered staging of row PAIRS (4 x 512 B), prefetch depth 3
    __shared__ __align__(16) float buf[4][2 * DP_J];

    stage_pair(gbase +   0 + 4 * lane, lds_byte_addr(&buf[0][4 * lane]));
    stage_pair(gbase + 128 + 4 * lane, lds_byte_addr(&buf[1][4 * lane]));
    stage_pair(gbase + 256 + 4 * lane, lds_byte_addr(&buf[2][4 * lane]));

    float v0 = 0.f, v1 = 0.f;

    // pair 0 (rows 0,1) peeled: row 0 is just the running sum (v[0]=0)
    wait_async<2>();
    {
        float2 wa = *(const float2*)(&buf[0][2 * lane]);
        float2 wb = *(const float2*)(&buf[0][DP_J + 2 * lane]);
        RowC ra = row_prefix(wa, lane, hi);
        RowC rb = row_prefix(wb, lane, hi);
        v0 = ra.c0; v1 = ra.c1;             // row 0: v = c
        row_combine(rb, lane, hi, v0, v1);  // row 1
    }

    // steady state: branch-free; always 4 pairs in flight
    for (int p = 1; p <= 28; ++p) {
        stage_pair(gbase + (size_t)(p + 3) * 128 + 4 * lane,
                   lds_byte_addr(&buf[(p + 3) & 3][4 * lane]));
        wait_async<3>();                    // outstanding: pairs p..p+3
        dp_pair(&buf[p & 3][0], lane, hi, v0, v1);
    }

    // drain epilogue: pairs 29, 30, 31 already in flight
    wait_async<2>();
    dp_pair(&buf[29 & 3][0], lane, hi, v0, v1);
    wait_async<1>();
    dp_pair(&buf[30 & 3][0], lane, hi, v0, v1);
    wait_async<0>();
    dp_pair(&buf[31 & 3][0], lane, hi, v0, v1);

    if (lane == 31) out[b] = v1;            // v[63,63]
}

extern "C" void kernel_launch(void* const* d_in, const int* in_sizes, int n_in,
                              void* d_out, int out_size, void* d_ws, size_t ws_size,
                              hipStream_t stream) {
    (void)in_sizes; (void)n_in; (void)d_ws; (void)ws_size;
    const float* img = (const float*)d_in[0];
    float* out = (float*)d_out;
    const int B = out_size;                 // one output per image
    if (B <= 0) return;
    grid_dp_kernel<<<B, 32, 0, stream>>>(img, out);
}